// GraphAttentionLineage_68358699483903
// MI455X (gfx1250) — compile-verified
//
#include <hip/hip_runtime.h>

typedef __attribute__((ext_vector_type(2))) float v2f;
typedef __attribute__((ext_vector_type(8))) float v8f;
typedef __attribute__((ext_vector_type(4))) unsigned int u32x4;
typedef __attribute__((ext_vector_type(8))) int i32x8;
typedef __attribute__((ext_vector_type(4))) int i32x4;

#define NEG_FLT_MAX (-3.402823466e+38f)
#define NEG_SLOPE 0.2f
#define LN_EPS 1e-5f

// ---------------------------------------------------------------------------
// TDM: DMA a 2D tile (tile_d1 rows x tile_d0 elems of f32) from global to LDS,
// inserting pad_amount DWORDs of padding every pad_interval DWORDs (codes per
// ISA 8.4: interval code 6 = 128 DWORDs, amount code 3 = 4 DWORDs).
// D# group0: count=1, lds_addr, global_addr(57b), type=2.
// D# group1: data_size=4B, pad fields, tensor dims, tile dims, dim0 stride.
// Groups 2/3 zero (2D tensor). Tracked by TENSORcnt.
// This toolchain uses the 6-arg builtin: (u32x4, i32x8, i32x4, i32x4, i32x8, cpol).
// ---------------------------------------------------------------------------
__device__ __forceinline__ void tdm_load_2d_f32(
    unsigned lds_addr, const float* gaddr,
    unsigned tensor_d0, unsigned tensor_d1,
    unsigned tile_d0, unsigned tile_d1,
    unsigned stride0_elems,
    unsigned pad_interval_code, unsigned pad_amount_code)
{
    unsigned long long ga = (unsigned long long)(uintptr_t)gaddr;
    u32x4 g0;
    g0[0] = 1u;                                   // count = 1 valid descriptor
    g0[1] = lds_addr;                             // LDS byte address
    g0[2] = (unsigned)(ga & 0xFFFFFFFFu);         // global_addr[31:0]
    g0[3] = (unsigned)((ga >> 32) & 0x1FFFFFFu)   // global_addr[56:32]
          | 0x80000000u;                          // type = 2 ("image")
    i32x8 g1;
    g1[0] = (int)((2u << 16)                      // data_size = 2 -> 4 bytes
          | (1u << 20)                            // pad_enable
          | (pad_interval_code << 22)
          | (pad_amount_code << 25));
    g1[1] = (int)((tensor_d0 & 0xFFFFu) << 16);   // tensor_dim0[15:0]
    g1[2] = (int)(((tensor_d0 >> 16) & 0xFFFFu)   // tensor_dim0[31:16]
          | ((tensor_d1 & 0xFFFFu) << 16));       // tensor_dim1[15:0]
    g1[3] = (int)(((tensor_d1 >> 16) & 0xFFFFu)   // tensor_dim1[31:16]
          | ((tile_d0 & 0xFFFFu) << 16));         // tile_dim0
    g1[4] = (int)(tile_d1 & 0xFFFFu);             // tile_dim1 (tile_dim2 = 0)
    g1[5] = (int)stride0_elems;                   // tensor_dim0_stride[31:0]
    g1[6] = 0;                                    // stride[47:32]=0, dim1_stride lo=0
    g1[7] = 0;
    i32x4 gz4 = {0, 0, 0, 0};
    i32x8 gz8 = {0, 0, 0, 0, 0, 0, 0, 0};
    __builtin_amdgcn_tensor_load_to_lds(g0, g1, gz4, gz4, gz8, 0);
}

// ---------------------------------------------------------------------------
// GEMM: out[M,128] = act(X[M,128] @ W[128,128] + bias)
// block = 256 threads = 8 waves; block covers 16 rows; wave w covers cols 16w..16w+15
// A tile staged into padded LDS by the Tensor Data Mover (wave 0 issues it);
// 32x V_WMMA_F32_16X16X4_F32 over K=128.
// ---------------------------------------------------------------------------
__global__ __launch_bounds__(256) void gemm128_wmma(
    const float* __restrict__ X, const float* __restrict__ W,
    const float* __restrict__ bias, float* __restrict__ out,
    int M, int act)
{
    __shared__ float As[16][132];  // 128 + 4 pad floats per row (bank-conflict free)
    const int row0 = blockIdx.x * 16;

    if ((threadIdx.x >> 5) == 0) {
        unsigned lds = (unsigned)(uintptr_t)(&As[0][0]);
        // 16x128 f32 tile; rows beyond M read as zero (TDM OOB -> 0).
        tdm_load_2d_f32(lds, X + (size_t)row0 * 128,
                        /*tensor_d0=*/128u, /*tensor_d1=*/(unsigned)(M - row0),
                        /*tile_d0=*/128u, /*tile_d1=*/16u,
                        /*stride0=*/128u,
                        /*pad_interval: 128 DW=*/6u, /*pad_amount: 4 DW=*/3u);
        __builtin_amdgcn_s_wait_tensorcnt(0);
    }
    __syncthreads();

    const int wave  = threadIdx.x >> 5;
    const int lane  = threadIdx.x & 31;
    const int lm    = lane & 15;
    const int khalf = (lane >> 4) << 1;   // lanes 0-15 -> K+0/K+1, lanes 16-31 -> K+2/K+3
    const int c     = wave * 16 + lm;

    v8f acc = {};
    #pragma unroll
    for (int k = 0; k < 128; k += 4) {
        v2f a, b;
        a[0] = As[lm][k + khalf];
        a[1] = As[lm][k + khalf + 1];
        b[0] = W[(size_t)(k + khalf) * 128 + c];
        b[1] = W[(size_t)(k + khalf + 1) * 128 + c];
        acc = __builtin_amdgcn_wmma_f32_16x16x4_f32(
            false, a, false, b, (short)0, acc, false, false);
    }

    const float bv = bias ? bias[c] : 0.0f;
    const int rbase = row0 + ((lane >> 4) << 3);  // lanes 16-31 hold M=j+8
    #pragma unroll
    for (int j = 0; j < 8; ++j) {
        int r = rbase + j;
        if (r < M) {
            float v = acc[j] + bv;
            if (act) v = fmaxf(v, 0.0f);
            out[(size_t)r * 128 + c] = v;
        }
    }
}

// ---------------------------------------------------------------------------
// al_s[n,h] = sum_c hW[n,h,c]*a_src[h,c]; al_d likewise. One wave per node.
// ---------------------------------------------------------------------------
__global__ __launch_bounds__(256) void attn_scores(
    const float* __restrict__ hW, const float* __restrict__ a_s,
    const float* __restrict__ a_d, float* __restrict__ al_s,
    float* __restrict__ al_d, int N, int H)
{
    int node = blockIdx.x * 8 + (threadIdx.x >> 5);
    if (node >= N) return;
    int lane = threadIdx.x & 31;

    float4 hv = *(const float4*)(hW + (size_t)node * 128 + lane * 4);
    float4 as = *(const float4*)(a_s + lane * 4);
    float4 ad = *(const float4*)(a_d + lane * 4);

    float ss = hv.x * as.x + hv.y * as.y + hv.z * as.z + hv.w * as.w;
    float sd = hv.x * ad.x + hv.y * ad.y + hv.z * ad.z + hv.w * ad.w;

    int lph = 32 / H;  // lanes per head (8 for H=4, 32 for H=1)
    for (int off = lph >> 1; off > 0; off >>= 1) {
        ss += __shfl_xor(ss, off, 32);
        sd += __shfl_xor(sd, off, 32);
    }
    if ((lane & (lph - 1)) == 0) {
        int h = lane / lph;
        al_s[(size_t)node * H + h] = ss;
        al_d[(size_t)node * H + h] = sd;
    }
}

// ---------------------------------------------------------------------------
__global__ void fill_f32(float* __restrict__ p, float v, int n)
{
    int i = blockIdx.x * blockDim.x + threadIdx.x;
    if (i < n) p[i] = v;
}

// float atomic max via signed-max / unsigned-min bit trick (init -FLT_MAX)
__device__ __forceinline__ void atomic_max_float(float* addr, float val)
{
    if (val >= 0.0f) atomicMax((int*)addr, __float_as_int(val));
    else             atomicMin((unsigned int*)addr, __float_as_uint(val));
}

__device__ __forceinline__ void edge_endpoints(const int* ei, int E, int e,
                                               int& s, int& d)
{
    if (e < E) { s = ei[e]; d = ei[E + e]; }
    else       { s = e - E; d = e - E; }   // self loop
}

// Pass 1: segment max of leaky_relu(al_s[src]+al_d[dst]) into m[dst,h]
__global__ void edge_max_kernel(
    const int* __restrict__ ei, int E, int N,
    const float* __restrict__ al_s, const float* __restrict__ al_d,
    float* __restrict__ m, int H)
{
    int idx = blockIdx.x * blockDim.x + threadIdx.x;
    int tot = (E + N) * H;
    if (idx >= tot) return;
    int e = idx / H, h = idx - e * H;
    int s, d; edge_endpoints(ei, E, e, s, d);
    float v = al_s[(size_t)s * H + h] + al_d[(size_t)d * H + h];
    v = (v > 0.0f) ? v : NEG_SLOPE * v;
    atomic_max_float(&m[(size_t)d * H + h], v);
}

// Pass 2: z[dst,h] += exp(e - m[dst,h])
__global__ void edge_sum_kernel(
    const int* __restrict__ ei, int E, int N,
    const float* __restrict__ al_s, const float* __restrict__ al_d,
    const float* __restrict__ m, float* __restrict__ z, int H)
{
    int idx = blockIdx.x * blockDim.x + threadIdx.x;
    int tot = (E + N) * H;
    if (idx >= tot) return;
    int e = idx / H, h = idx - e * H;
    int s, d; edge_endpoints(ei, E, e, s, d);
    float v = al_s[(size_t)s * H + h] + al_d[(size_t)d * H + h];
    v = (v > 0.0f) ? v : NEG_SLOPE * v;
    float ex = expf(v - m[(size_t)d * H + h]);
    atomicAdd(&z[(size_t)d * H + h], ex);
}

// Pass 3: agg[dst,:] += hW[src,:] * alpha  (one wave per edge, float4 per lane)
__global__ __launch_bounds__(256) void edge_aggregate(
    const int* __restrict__ ei, int E, int N,
    const float* __restrict__ al_s, const float* __restrict__ al_d,
    const float* __restrict__ m, const float* __restrict__ z,
    const float* __restrict__ hW, float* __restrict__ agg, int H)
{
    int e = blockIdx.x * 8 + (threadIdx.x >> 5);
    if (e >= E + N) return;
    int lane = threadIdx.x & 31;
    int s, d; edge_endpoints(ei, E, e, s, d);

    int h = (lane * H) >> 5;  // head owning features [lane*4 .. lane*4+3]
    float v = al_s[(size_t)s * H + h] + al_d[(size_t)d * H + h];
    v = (v > 0.0f) ? v : NEG_SLOPE * v;
    float alpha = expf(v - m[(size_t)d * H + h]) /
                  (z[(size_t)d * H + h] + 1e-16f);

    float4 hv = *(const float4*)(hW + (size_t)s * 128 + lane * 4);
    float* dp = agg + (size_t)d * 128 + lane * 4;
    atomicAdd(dp + 0, hv.x * alpha);
    atomicAdd(dp + 1, hv.y * alpha);
    atomicAdd(dp + 2, hv.z * alpha);
    atomicAdd(dp + 3, hv.w * alpha);
}

// ---------------------------------------------------------------------------
// post-processing: mode 1: out = LN(elu(agg+bias) + res)   mode 0: out = LN(agg+bias)
// one wave per node; LN over 128 features (population variance)
// ---------------------------------------------------------------------------
__global__ __launch_bounds__(256) void post_ln(
    const float* __restrict__ agg, const float* __restrict__ bias,
    const float* __restrict__ res, float* __restrict__ out,
    int N, int do_elu_res)
{
    int node = blockIdx.x * 8 + (threadIdx.x >> 5);
    if (node >= N) return;
    int lane = threadIdx.x & 31;

    float4 a = *(const float4*)(agg + (size_t)node * 128 + lane * 4);
    float4 b = *(const float4*)(bias + lane * 4);
    float vx = a.x + b.x, vy = a.y + b.y, vz = a.z + b.z, vw = a.w + b.w;

    if (do_elu_res) {
        float4 r = *(const float4*)(res + (size_t)node * 128 + lane * 4);
        vx = ((vx > 0.0f) ? vx : (expf(vx) - 1.0f)) + r.x;
        vy = ((vy > 0.0f) ? vy : (expf(vy) - 1.0f)) + r.y;
        vz = ((vz > 0.0f) ? vz : (expf(vz) - 1.0f)) + r.z;
        vw = ((vw > 0.0f) ? vw : (expf(vw) - 1.0f)) + r.w;
    }

    float sum = vx + vy + vz + vw;
    float sq  = vx * vx + vy * vy + vz * vz + vw * vw;
    for (int off = 16; off > 0; off >>= 1) {
        sum += __shfl_xor(sum, off, 32);
        sq  += __shfl_xor(sq,  off, 32);
    }
    float mean = sum * (1.0f / 128.0f);
    float var  = sq * (1.0f / 128.0f) - mean * mean;
    float inv  = rsqrtf(var + LN_EPS);

    float4 o;
    o.x = (vx - mean) * inv;
    o.y = (vy - mean) * inv;
    o.z = (vz - mean) * inv;
    o.w = (vw - mean) * inv;
    *(float4*)(out + (size_t)node * 128 + lane * 4) = o;
}

// ---------------------------------------------------------------------------
extern "C" void kernel_launch(void* const* d_in, const int* in_sizes, int n_in,
                              void* d_out, int out_size, void* d_ws, size_t ws_size,
                              hipStream_t stream)
{
    const float* x    = (const float*)d_in[0];
    const int*   ei   = (const int*)  d_in[1];
    const float* W_in = (const float*)d_in[2];
    const float* b_in = (const float*)d_in[3];
    const float* W0   = (const float*)d_in[4];
    const float* as0  = (const float*)d_in[5];
    const float* ad0  = (const float*)d_in[6];
    const float* b0   = (const float*)d_in[7];
    const float* W1   = (const float*)d_in[8];
    const float* as1  = (const float*)d_in[9];
    const float* ad1  = (const float*)d_in[10];
    const float* b1   = (const float*)d_in[11];
    const float* W2   = (const float*)d_in[12];
    const float* as2  = (const float*)d_in[13];
    const float* ad2  = (const float*)d_in[14];
    const float* b2   = (const float*)d_in[15];

    const int N = in_sizes[0] / 128;
    const int E = in_sizes[1] / 2;
    const int Etot = E + N;

    float* ws   = (float*)d_ws;
    float* h    = ws; ws += (size_t)N * 128;
    float* hW   = ws; ws += (size_t)N * 128;
    float* agg  = ws; ws += (size_t)N * 128;
    float* al_s = ws; ws += (size_t)N * 4;
    float* al_d = ws; ws += (size_t)N * 4;
    float* mbuf = ws; ws += (size_t)N * 4;
    float* zbuf = ws; ws += (size_t)N * 4;

    const dim3 blk(256);
    const int gemm_grid = (N + 15) / 16;
    const int node_grid = (N + 7) / 8;
    const int edge_wave_grid = (Etot + 7) / 8;

    // h = relu(x @ W_in + b_in)
    gemm128_wmma<<<gemm_grid, blk, 0, stream>>>(x, W_in, b_in, h, N, 1);

    const float* Ws[2]  = {W0, W1};
    const float* ass[2] = {as0, as1};
    const float* ads[2] = {ad0, ad1};
    const float* bs[2]  = {b0, b1};

    for (int l = 0; l < 2; ++l) {
        const int H = 4;
        gemm128_wmma<<<gemm_grid, blk, 0, stream>>>(h, Ws[l], nullptr, hW, N, 0);
        attn_scores<<<node_grid, blk, 0, stream>>>(hW, ass[l], ads[l], al_s, al_d, N, H);
        fill_f32<<<(N * H + 255) / 256, blk, 0, stream>>>(mbuf, NEG_FLT_MAX, N * H);
        fill_f32<<<(N * H + 255) / 256, blk, 0, stream>>>(zbuf, 0.0f, N * H);
        fill_f32<<<(N * 128 + 255) / 256, blk, 0, stream>>>(agg, 0.0f, N * 128);
        int tg = (Etot * H + 255) / 256;
        edge_max_kernel<<<tg, blk, 0, stream>>>(ei, E, N, al_s, al_d, mbuf, H);
        edge_sum_kernel<<<tg, blk, 0, stream>>>(ei, E, N, al_s, al_d, mbuf, zbuf, H);
        edge_aggregate<<<edge_wave_grid, blk, 0, stream>>>(ei, E, N, al_s, al_d,
                                                           mbuf, zbuf, hW, agg, H);
        // h = LN(elu(agg + b) + h)   (in-place on h is safe: wave-private node)
        post_ln<<<node_grid, blk, 0, stream>>>(agg, bs[l], h, h, N, 1);
    }

    // Layer 2: heads = 1, no concat (mean over 1 head == identity)
    {
        const int H = 1;
        gemm128_wmma<<<gemm_grid, blk, 0, stream>>>(h, W2, nullptr, hW, N, 0);
        attn_scores<<<node_grid, blk, 0, stream>>>(hW, as2, ad2, al_s, al_d, N, H);
        fill_f32<<<(N * H + 255) / 256, blk, 0, stream>>>(mbuf, NEG_FLT_MAX, N * H);
        fill_f32<<<(N * H + 255) / 256, blk, 0, stream>>>(zbuf, 0.0f, N * H);
        fill_f32<<<(N * 128 + 255) / 256, blk, 0, stream>>>(agg, 0.0f, N * 128);
        int tg = (Etot * H + 255) / 256;
        edge_max_kernel<<<tg, blk, 0, stream>>>(ei, E, N, al_s, al_d, mbuf, H);
        edge_sum_kernel<<<tg, blk, 0, stream>>>(ei, E, N, al_s, al_d, mbuf, zbuf, H);
        edge_aggregate<<<edge_wave_grid, blk, 0, stream>>>(ei, E, N, al_s, al_d,
                                                           mbuf, zbuf, hW, agg, H);
        // out = LN(agg + b2)
        post_ln<<<node_grid, blk, 0, stream>>>(agg, b2, nullptr, (float*)d_out, N, 0);
    }
}